// BaseModel_9277129359377
// MI455X (gfx1250) — compile-verified
//
#include <hip/hip_runtime.h>
#include <stdint.h>

typedef float v2f __attribute__((ext_vector_type(2)));
typedef float v8f __attribute__((ext_vector_type(8)));

#if defined(__HIP_DEVICE_COMPILE__)
#if !__has_builtin(__builtin_amdgcn_wmma_f32_16x16x4_f32)
#error "missing __builtin_amdgcn_wmma_f32_16x16x4_f32 (device)"
#endif
#endif

#define S_TOTAL 400000
#define WIN     400
#define NWIN    1000
#define NR      64
#define NA      3
#define NB      2
#define KTOP    8
#define KCH     100
#define NCHUNK  (WIN / KCH)
#define CHUNK_V4 (16 * (KCH / 4))   // 400 float4 per wave-chunk -> 13 async ops/wave

// Async global->LDS copy (CDNA5): data bypasses VGPRs, tracked by ASYNCcnt.
__device__ __forceinline__ void async_ld_b128(uint32_t lds_off, const float* g) {
    asm volatile("global_load_async_to_lds_b128 %0, %1, off"
                 :: "v"(lds_off), "v"((uint64_t)(uintptr_t)g)
                 : "memory");
}
#define WAIT_ASYNC_13() asm volatile("s_wait_asynccnt 0xd" ::: "memory")
#define WAIT_ASYNC_0()  asm volatile("s_wait_asynccnt 0x0" ::: "memory")

// One block per (a, w): 4 waves, wave i owns ref rows [16i, 16i+16).
// C[r,b] = (1/WIN) * sum_k ref[a,r,w*WIN+k] * mixed[b,w*WIN+k] via
// v_wmma_f32_16x16x4_f32 (A = ref rows, B = mixed columns, only n=0,1 used).
// Double-buffered: chunk c+1 streams into LDS (async) while WMMAs run on c.
__global__ __launch_bounds__(128) void windowed_topk_kernel(
    const float* __restrict__ mixed,
    const float* __restrict__ ref,
    const float* __restrict__ weights,
    float* __restrict__ outv,
    int* __restrict__ outi)
{
    __shared__ __align__(16) float ldsRef[2][4][16][KCH];  // 51.2 KB (double buffer)
    __shared__ __align__(16) float ldsMix[NB][WIN];        // 3.2 KB
    __shared__ float scores[NB][NR];

    const int tid  = threadIdx.x;
    const int wv   = tid >> 5;
    const int lane = tid & 31;
    const int a    = blockIdx.x / NWIN;
    const int w    = blockIdx.x % NWIN;

    const float* refBase = ref + ((size_t)a * NR) * S_TOTAL + (size_t)w * WIN;
    const int rbase = wv * 16;

    // Stage this window of `mixed` for both batches (16B-aligned float4 loads).
    for (int t = tid; t < NB * (WIN / 4); t += blockDim.x) {
        const int b = t / (WIN / 4);
        const int q = t % (WIN / 4);
        const float4 v = *(const float4*)(mixed + (size_t)b * S_TOTAL + (size_t)w * WIN + q * 4);
        *(float4*)&ldsMix[b][q * 4] = v;
    }

    v8f acc = {0.f, 0.f, 0.f, 0.f, 0.f, 0.f, 0.f, 0.f};

    const int half  = lane >> 4;      // 0: K offsets {0,1}; 1: K offsets {2,3}
    const int mrow  = lane & 15;      // A-matrix M = local ref row; B-matrix N
    const bool nvalid = (mrow < NB);

#if defined(__HIP_DEVICE_COMPILE__)
    // Prologue: async-issue chunk 0 into buffer 0 (13 b128 ops per wave).
    {
        float* buf = &ldsRef[0][wv][0][0];
        for (int t = lane; t < CHUNK_V4; t += 32) {
            const int row = t / (KCH / 4);
            const int q   = t % (KCH / 4);
            async_ld_b128((uint32_t)(uintptr_t)(buf + row * KCH + q * 4),
                          refBase + (size_t)(rbase + row) * S_TOTAL + q * 4);
        }
    }
    __syncthreads();   // ldsMix visible to all waves before compute

    for (int c = 0; c < NCHUNK; ++c) {
        // Issue next chunk into the other buffer, then wait for this chunk.
        if (c + 1 < NCHUNK) {
            float* buf = &ldsRef[(c + 1) & 1][wv][0][0];
            for (int t = lane; t < CHUNK_V4; t += 32) {
                const int row = t / (KCH / 4);
                const int q   = t % (KCH / 4);
                async_ld_b128((uint32_t)(uintptr_t)(buf + row * KCH + q * 4),
                              refBase + (size_t)(rbase + row) * S_TOTAL + (c + 1) * KCH + q * 4);
            }
            WAIT_ASYNC_13();   // <=13 outstanding => chunk c fully landed
        } else {
            WAIT_ASYNC_0();
        }

        const float* rbuf = &ldsRef[c & 1][wv][mrow][0];
        #pragma unroll
        for (int koff = 0; koff < KCH; koff += 4) {
            const int kk = koff + 2 * half;
            // A: ref[a, rbase+mrow, base + kk .. kk+1]  (documented A layout)
            const v2f aF = *(const v2f*)&rbuf[kk];
            // B: mixed[n, base + kk .. kk+1], n = mrow; zero for n >= 2
            const v2f bv = *(const v2f*)&ldsMix[mrow & 1][c * KCH + kk];
            v2f bF;
            bF.x = nvalid ? bv.x : 0.0f;
            bF.y = nvalid ? bv.y : 0.0f;
            acc = __builtin_amdgcn_wmma_f32_16x16x4_f32(
                false, aF, false, bF, (short)0, acc, false, false);
        }
    }
#endif

    // D layout: VGPR v holds M=v (lanes 0-15, N=lane) and M=v+8 (lanes 16-31).
    #pragma unroll
    for (int v = 0; v < 8; ++v) {
        const float s = acc[v] * (1.0f / (float)WIN);
        if (lane < NB)                         scores[lane][rbase + v]          = s;
        else if (lane >= 16 && lane < 16 + NB) scores[lane - 16][rbase + 8 + v] = s;
    }
    __syncthreads();

    // Stable top-8 over the 64 refs (lower index wins ties, like jax top_k).
    if (tid < NB) {
        const int b = tid;
        float vals[KTOP];
        int   ids[KTOP];
        unsigned long long used = 0ull;
        for (int k = 0; k < KTOP; ++k) {
            float best = -3.402823466e38f;
            int   bi   = 0;
            for (int r = 0; r < NR; ++r) {
                if (!((used >> r) & 1ull)) {
                    const float x = scores[b][r];
                    if (x > best) { best = x; bi = r; }
                }
            }
            used |= (1ull << bi);
            vals[k] = best;
            ids[k]  = bi;
        }
        float o = 0.0f;
        #pragma unroll
        for (int k = 0; k < KTOP; ++k) o += vals[k] * weights[k];
        const int ow = (b * NA + a) * NWIN + w;
        outv[ow] = o;
        #pragma unroll
        for (int k = 0; k < KTOP; ++k) outi[ow * KTOP + k] = ids[k];
    }
}

extern "C" void kernel_launch(void* const* d_in, const int* in_sizes, int n_in,
                              void* d_out, int out_size, void* d_ws, size_t ws_size,
                              hipStream_t stream) {
    const float* mixed   = (const float*)d_in[0];   // [2, 400000]
    const float* ref     = (const float*)d_in[1];   // [3, 64, 400000]
    const float* weights = (const float*)d_in[2];   // [8, 1]
    float* outv = (float*)d_out;                    // [2,3,1000] f32
    int*   outi = (int*)d_out + NB * NA * NWIN;     // [2,3,1000,8] i32

    dim3 grid(NA * NWIN);   // 3000 blocks, one per (a, w)
    windowed_topk_kernel<<<grid, 128, 0, stream>>>(mixed, ref, weights, outv, outi);
}